// LaplacianBlurDetector_83674552860692
// MI455X (gfx1250) — compile-verified
//
#include <hip/hip_runtime.h>
#include <cstdint>
#include <cstddef>

#define NBINS 256
#define IMG_W 1024
#define IMG_H 1024
#define NIMG  64

// ---- tile geometry for the Laplacian pass ----
#define TW 256                 // tile width in pixels
#define THT 32                 // tile height in pixels
#define TROWS (THT + 2)        // 34 rows incl. vertical halo
#define HALOL 16               // left/right halo bytes (16B-aligned async b128)
#define TSTRIDE (TW + 2 * HALOL)   // 288 bytes/row, 288 % 16 == 0
#define TWORDS (TROWS * TSTRIDE / 4)

#define AS1 __attribute__((address_space(1)))
#define AS3 __attribute__((address_space(3)))

typedef int v4i_t __attribute__((ext_vector_type(4)));

#if __has_builtin(__builtin_amdgcn_global_load_async_to_lds_b32)
#define ASYNC_LDS_B32 1
#else
#define ASYNC_LDS_B32 0
#endif
#if __has_builtin(__builtin_amdgcn_global_load_async_to_lds_b128)
#define ASYNC_LDS_B128 1
#else
#define ASYNC_LDS_B128 0
#endif

__device__ __forceinline__ void async_load_b32(const uint8_t* gp, uint8_t* lp) {
#if ASYNC_LDS_B32
  AS3 uint8_t* l3 = (AS3 uint8_t*)lp;            // addrspacecast generic -> LDS
  __builtin_amdgcn_global_load_async_to_lds_b32(
      (AS1 int*)(unsigned long long)gp, (AS3 int*)l3, 0, 0);
#else
  unsigned loff = (unsigned)(unsigned long long)(AS3 uint8_t*)lp;
  asm volatile("global_load_async_to_lds_b32 %0, %1, off"
               :: "v"(loff), "v"(gp) : "memory");
#endif
}

__device__ __forceinline__ void async_load_b128(const uint8_t* gp, uint8_t* lp) {
#if ASYNC_LDS_B128
  AS3 uint8_t* l3 = (AS3 uint8_t*)lp;
  __builtin_amdgcn_global_load_async_to_lds_b128(
      (AS1 v4i_t*)(unsigned long long)gp, (AS3 v4i_t*)l3, 0, 0);
#else
  async_load_b32(gp, lp);
  async_load_b32(gp + 4, lp + 4);
  async_load_b32(gp + 8, lp + 8);
  async_load_b32(gp + 12, lp + 12);
#endif
}

__device__ __forceinline__ void wait_async0() {
#if __has_builtin(__builtin_amdgcn_s_wait_asynccnt)
  __builtin_amdgcn_s_wait_asynccnt(0);
#else
  asm volatile("s_wait_asynccnt 0" ::: "memory");
#endif
}

// ---------------------------------------------------------------------------
// Pass 0: zero the per-image histograms (graph-capture-safe init)
// ---------------------------------------------------------------------------
__global__ __launch_bounds__(256) void zero_hist_kernel(unsigned* hist) {
  hist[blockIdx.x * NBINS + threadIdx.x] = 0u;
}

// ---------------------------------------------------------------------------
// Pass 1: per-image histogram (LDS ds_add_u32) + optional u8 compression
// grid: (64 chunks, 64 images), 256 threads; 16384 px per block
// ---------------------------------------------------------------------------
template <bool WRITE_U8>
__global__ __launch_bounds__(256) void hist_kernel(const float* __restrict__ img,
                                                   uint8_t* __restrict__ u8img,
                                                   unsigned* __restrict__ hist) {
  __shared__ unsigned lh[NBINS];
  const int t = threadIdx.x;
  lh[t] = 0u;
  __syncthreads();

  const int image = blockIdx.y;
  const size_t cbase = ((size_t)image << 20) + ((size_t)blockIdx.x << 14);
  const float4* __restrict__ src = (const float4*)(img + cbase);
  uint32_t* __restrict__ dst = (uint32_t*)(u8img + cbase);

#pragma unroll 4
  for (int i = 0; i < 16; ++i) {
    float4 f = src[i * 256 + t];
    int v0 = min(max((int)f.x, 0), 255);
    int v1 = min(max((int)f.y, 0), 255);
    int v2 = min(max((int)f.z, 0), 255);
    int v3 = min(max((int)f.w, 0), 255);
    atomicAdd(&lh[v0], 1u);
    atomicAdd(&lh[v1], 1u);
    atomicAdd(&lh[v2], 1u);
    atomicAdd(&lh[v3], 1u);
    if (WRITE_U8) {
      dst[i * 256 + t] = (unsigned)v0 | ((unsigned)v1 << 8) |
                         ((unsigned)v2 << 16) | ((unsigned)v3 << 24);
    }
  }
  __syncthreads();
  atomicAdd(&hist[image * NBINS + t], lh[t]);
}

// ---------------------------------------------------------------------------
// Pass 2: CDF scan + LUT (one block per image, 256 threads)
// ---------------------------------------------------------------------------
__global__ __launch_bounds__(256) void lut_kernel(const unsigned* __restrict__ hist,
                                                  float* __restrict__ lutf) {
  __shared__ float s[NBINS];
  __shared__ unsigned minidx;
  const int t = threadIdx.x;
  const int image = blockIdx.x;
  const unsigned h = hist[image * NBINS + t];
  s[t] = (float)h;                    // counts < 2^24: exact in f32 (matches JAX)
  if (t == 0) minidx = NBINS;
  __syncthreads();
  for (int o = 1; o < NBINS; o <<= 1) {   // Hillis-Steele inclusive scan
    float v = (t >= o) ? s[t - o] : 0.0f;
    __syncthreads();
    s[t] += v;
    __syncthreads();
  }
  if (h > 0u) atomicMin(&minidx, (unsigned)t);
  __syncthreads();
  const float cdfmin = s[minidx < NBINS ? minidx : 0];
  const float total = (float)(IMG_W * IMG_H);
  const float scale = 255.0f / fmaxf(total - cdfmin, 1.0f);
  float l = rintf((s[t] - cdfmin) * scale);   // RNE == jnp.round
  l = fminf(fmaxf(l, 0.0f), 255.0f);
  lutf[image * NBINS + t] = l * (1.0f / 255.0f);
}

// ---------------------------------------------------------------------------
// Pass 3: LUT apply + 3x3 Laplacian (zero pad) + per-block (sum, sumsq)
// grid: (4, 32, 64), 256 threads. Tile staged in LDS as u8 with halo.
// USE_U8: async b128 global->LDS copies from compressed image.
// else  : float loads from original image, converted in VGPRs, ds_store'd.
// ---------------------------------------------------------------------------
template <bool USE_U8>
__global__ __launch_bounds__(256) void lap_kernel(const float* __restrict__ fimg,
                                                  const uint8_t* __restrict__ u8img,
                                                  const float* __restrict__ lutf,
                                                  float* __restrict__ partials) {
  __shared__ __align__(16) uint8_t tile[TROWS * TSTRIDE];
  __shared__ float lut[NBINS];
  __shared__ float rs[256];
  __shared__ float rs2[256];

  const int t = threadIdx.x;
  const int image = blockIdx.z;
  const int gx0 = blockIdx.x * TW;
  const int gy0 = blockIdx.y * THT;

  lut[t] = lutf[image * NBINS + t];
  // zero-fill tile => halo outside the image is exact zero padding
  uint32_t* tw = (uint32_t*)tile;
  for (int i = t; i < TWORDS; i += 256) tw[i] = 0u;
  __syncthreads();

  // ---- stage the in-bounds tile region into LDS ----
  const int gyf = max(gy0 - 1, 0);
  const int gyl = min(gy0 + THT, IMG_H - 1);
  const int nrows = gyl - gyf + 1;
  const size_t ib = (size_t)image << 20;

  if (USE_U8) {
    // 16B-aligned chunks: gx0 is a multiple of 256, so gx0-16 is 16B aligned,
    // and LDS column (gx - (gx0-16)) keeps 16B alignment (TSTRIDE % 16 == 0).
    const int xs = max(gx0 - HALOL, 0);
    const int xe = min(gx0 + TW + HALOL, IMG_W);
    const int nch = (xe - xs) >> 4;
    const int total = nrows * nch;
    for (int idx = t; idx < total; idx += 256) {
      const int r = idx / nch;
      const int c = idx - r * nch;
      const int gy = gyf + r;
      const int gx = xs + (c << 4);
      uint8_t* lp = &tile[(gy - (gy0 - 1)) * TSTRIDE + (gx - (gx0 - HALOL))];
      const uint8_t* gp = u8img + ib + ((size_t)gy << 10) + gx;
      async_load_b128(gp, lp);
    }
    wait_async0();
  } else {
    const int xs = max(gx0 - 4, 0);
    const int xe = min(gx0 + TW + 4, IMG_W);
    const int nwords = (xe - xs) >> 2;
    const int total = nrows * nwords;
    for (int idx = t; idx < total; idx += 256) {
      const int r = idx / nwords;
      const int w = idx - r * nwords;
      const int gy = gyf + r;
      const int gx = xs + (w << 2);
      uint8_t* lp = &tile[(gy - (gy0 - 1)) * TSTRIDE + (gx - gx0 + HALOL)];
      const float* gp = fimg + ib + ((size_t)gy << 10) + gx;
      float4 f = *(const float4*)gp;
      int v0 = min(max((int)f.x, 0), 255);
      int v1 = min(max((int)f.y, 0), 255);
      int v2 = min(max((int)f.z, 0), 255);
      int v3 = min(max((int)f.w, 0), 255);
      *(uint32_t*)lp = (unsigned)v0 | ((unsigned)v1 << 8) |
                       ((unsigned)v2 << 16) | ((unsigned)v3 << 24);
    }
  }
  __syncthreads();

  // ---- compute: thread handles 4 consecutive x, 8 rows ----
  const int wx = t & 63;       // word column
  const int ty = t >> 6;       // row-group 0..3
  const int c0 = HALOL + (wx << 2);
  const int y0 = ty * (THT / 4);

  float s = 0.0f, s2 = 0.0f;
  float eT[4], eC[4], eB[4];
  uint32_t w0 = *(const uint32_t*)&tile[(y0)*TSTRIDE + c0];
  uint32_t w1 = *(const uint32_t*)&tile[(y0 + 1) * TSTRIDE + c0];
#pragma unroll
  for (int j = 0; j < 4; ++j) {
    eT[j] = lut[(w0 >> (8 * j)) & 255];
    eC[j] = lut[(w1 >> (8 * j)) & 255];
  }
  for (int y = y0; y < y0 + THT / 4; ++y) {
    uint32_t wB = *(const uint32_t*)&tile[(y + 2) * TSTRIDE + c0];
    unsigned bl = tile[(y + 1) * TSTRIDE + c0 - 1];
    unsigned br = tile[(y + 1) * TSTRIDE + c0 + 4];
#pragma unroll
    for (int j = 0; j < 4; ++j) eB[j] = lut[(wB >> (8 * j)) & 255];
    const float eL = lut[bl];
    const float eR = lut[br];
#pragma unroll
    for (int j = 0; j < 4; ++j) {
      const float left = (j == 0) ? eL : eC[j - 1];
      const float right = (j == 3) ? eR : eC[j + 1];
      const float lap = eT[j] + eB[j] + left + right - 4.0f * eC[j];
      s += lap;
      s2 = fmaf(lap, lap, s2);
    }
#pragma unroll
    for (int j = 0; j < 4; ++j) { eT[j] = eC[j]; eC[j] = eB[j]; }
  }

  rs[t] = s;
  rs2[t] = s2;
  __syncthreads();
  for (int o = 128; o > 0; o >>= 1) {
    if (t < o) { rs[t] += rs[t + o]; rs2[t] += rs2[t + o]; }
    __syncthreads();
  }
  if (t == 0) {
    const int bid = (blockIdx.z * gridDim.y + blockIdx.y) * gridDim.x + blockIdx.x;
    partials[2 * bid] = rs[0];
    partials[2 * bid + 1] = rs2[0];
  }
}

// ---------------------------------------------------------------------------
// Pass 4: final double-precision reduction -> unbiased variance
// ---------------------------------------------------------------------------
__global__ __launch_bounds__(256) void finish_kernel(const float* __restrict__ partials,
                                                     int n, float* __restrict__ out) {
  __shared__ double ds[256];
  __shared__ double ds2[256];
  const int t = threadIdx.x;
  double a = 0.0, b = 0.0;
  for (int i = t; i < n; i += 256) {
    a += (double)partials[2 * i];
    b += (double)partials[2 * i + 1];
  }
  ds[t] = a;
  ds2[t] = b;
  __syncthreads();
  for (int o = 128; o > 0; o >>= 1) {
    if (t < o) { ds[t] += ds[t + o]; ds2[t] += ds2[t + o]; }
    __syncthreads();
  }
  if (t == 0) {
    const double N = (double)NIMG * IMG_W * IMG_H;
    const double S = ds[0], S2 = ds2[0];
    out[0] = (float)((S2 - S * S / N) / (N - 1.0));
  }
}

// ---------------------------------------------------------------------------
extern "C" void kernel_launch(void* const* d_in, const int* in_sizes, int n_in,
                              void* d_out, int out_size, void* d_ws, size_t ws_size,
                              hipStream_t stream) {
  (void)in_sizes; (void)n_in; (void)out_size;
  const float* img = (const float*)d_in[0];   // [64,1,1024,1024] f32
  // d_in[1] is the fixed Laplacian kernel; hardcoded in lap_kernel.
  float* out = (float*)d_out;

  uint8_t* ws = (uint8_t*)d_ws;
  unsigned* hist = (unsigned*)(ws + 0);               //  64 KB
  float* lutf = (float*)(ws + 65536);                 //  64 KB
  float* partials = (float*)(ws + 131072);            //  64 KB
  uint8_t* u8img = ws + 196608;                       //  64 MB (optional)
  const size_t need = 196608 + (size_t)NIMG * IMG_W * IMG_H;
  const bool use_u8 = ws_size >= need;

  zero_hist_kernel<<<NIMG, 256, 0, stream>>>(hist);

  dim3 gA(64, NIMG);
  if (use_u8)
    hist_kernel<true><<<gA, 256, 0, stream>>>(img, u8img, hist);
  else
    hist_kernel<false><<<gA, 256, 0, stream>>>(img, u8img, hist);

  lut_kernel<<<NIMG, 256, 0, stream>>>(hist, lutf);

  dim3 gC(IMG_W / TW, IMG_H / THT, NIMG);             // (4, 32, 64)
  if (use_u8)
    lap_kernel<true><<<gC, 256, 0, stream>>>(img, u8img, lutf, partials);
  else
    lap_kernel<false><<<gC, 256, 0, stream>>>(img, u8img, lutf, partials);

  const int nblocks = (IMG_W / TW) * (IMG_H / THT) * NIMG;
  finish_kernel<<<1, 256, 0, stream>>>(partials, nblocks, out);
}